// CantorExpert_14903536517166
// MI455X (gfx1250) — compile-verified
//
#include <hip/hip_runtime.h>
#include <math.h>

// ---------------- problem constants (from reference) ----------------
#define FULL_DIM   8192
#define EXPERT_DIM 2048
#define SLICE      1024      // FULL_DIM / NUM_EXPERTS
#define SL_START   3072      // EXPERT_ID * SLICE
#define HID        256       // SLICE / 4
#define B_DIM      8
#define P_DIM      4096
// FP window for expert 3: [0.375, 0.5)

typedef __attribute__((ext_vector_type(2))) float v2f;
typedef __attribute__((ext_vector_type(8))) float v8f;
typedef __attribute__((ext_vector_type(4))) int   v4i;

// ---- CDNA5 async global->LDS staging (guarded; falls back if absent) ----
#if __has_builtin(__builtin_amdgcn_global_load_async_to_lds_b128) && \
    __has_builtin(__builtin_amdgcn_s_wait_asynccnt)
#define HAVE_ASYNC_LDS 1
#else
#define HAVE_ASYNC_LDS 0
#endif

#if HAVE_ASYNC_LDS
// clang signature (from probe diagnostics): (v4i* src, v4i* dst, Ii, Ii),
// generic pointers; backend resolves global/LDS address spaces.
#define GLDS_B128(g, l) \
  __builtin_amdgcn_global_load_async_to_lds_b128((v4i*)(g), (v4i*)(l), 0, 0)
#endif

// =====================================================================
// Kernel 1: fingerprint mask + index compaction (single block).
// =====================================================================
__global__ __launch_bounds__(1024) void k_mask_compact(
    const float* __restrict__ fp, int* __restrict__ wsN,
    int* __restrict__ wsIdx, float* __restrict__ dout)
{
  __shared__ int sc[1024];
  const int tid = threadIdx.x;
  int flags[4];
  int cnt = 0;
#pragma unroll
  for (int u = 0; u < 4; ++u) {
    float f = fp[tid * 4 + u];
    int m = (f >= 0.375f) && (f < 0.5f);
    flags[u] = m;
    cnt += m;
  }
  sc[tid] = cnt;
  __syncthreads();
  for (int off = 1; off < 1024; off <<= 1) {
    int v = (tid >= off) ? sc[tid - off] : 0;
    __syncthreads();
    sc[tid] += v;
    __syncthreads();
  }
  const int excl = sc[tid] - cnt;
  const int total = sc[1023];
  int w = excl;
#pragma unroll
  for (int u = 0; u < 4; ++u)
    if (flags[u]) wsIdx[w++] = tid * 4 + u;
  if (tid == 0) wsN[0] = total;
  const size_t MN = (size_t)B_DIM * (size_t)total;
  const size_t maskOff = MN * (size_t)(3 * EXPERT_DIM + 15);
#pragma unroll
  for (int u = 0; u < 4; ++u)
    dout[maskOff + tid * 4 + u] = flags[u] ? 1.0f : 0.0f;
}

// =====================================================================
// Kernel 2: normalize pentachoron directions into workspace
// =====================================================================
__global__ __launch_bounds__(256) void k_dirs(
    const float* __restrict__ pent, float* __restrict__ wsDirs)
{
  const int v = blockIdx.x;
  const int tid = threadIdx.x;
  __shared__ float red[256];
  float acc = 0.0f;
  for (int k = tid; k < EXPERT_DIM; k += 256) {
    float x = pent[v * EXPERT_DIM + k];
    acc += x * x;
  }
  red[tid] = acc;
  __syncthreads();
  for (int s = 128; s > 0; s >>= 1) {
    if (tid < s) red[tid] += red[tid + s];
    __syncthreads();
  }
  const float inv = 1.0f / fmaxf(sqrtf(red[0]), 1e-12f);
  for (int k = tid; k < EXPERT_DIM; k += 256)
    wsDirs[v * EXPERT_DIM + k] = pent[v * EXPERT_DIM + k] * inv;
}

// =====================================================================
// Kernel 3: gating MLP -> per-row scale + token-row base offset
// =====================================================================
__global__ __launch_bounds__(256) void k_gate(
    const float* __restrict__ tokens,
    const float* __restrict__ agw1, const float* __restrict__ agb1,
    const float* __restrict__ agw2, const float* __restrict__ agb2,
    const float* __restrict__ alpha,
    const int* __restrict__ wsN, const int* __restrict__ wsIdx,
    float* __restrict__ wsScale, unsigned long long* __restrict__ wsRowBase)
{
  const int N = wsN[0];
  const int j = blockIdx.x;
  if (j >= N) return;
  const int b = blockIdx.y;
  const int p = wsIdx[j];
  const size_t base = ((size_t)b * P_DIM + p) * FULL_DIM + SL_START;

  __shared__ float fLDS[SLICE];
  __shared__ float red[256];
  const int tid = threadIdx.x;
  for (int k = tid; k < SLICE; k += 256) fLDS[k] = tokens[base + k];
  __syncthreads();

  float acc = agb1[tid];
  for (int k = 0; k < SLICE; ++k)
    acc = fmaf(fLDS[k], agw1[(size_t)k * HID + tid], acc);
  const float g = 0.5f * acc * (1.0f + erff(acc * 0.70710678118654752f));
  red[tid] = g * agw2[tid];
  __syncthreads();
  for (int s = 128; s > 0; s >>= 1) {
    if (tid < s) red[tid] += red[tid + s];
    __syncthreads();
  }
  if (tid == 0) {
    const float zz = red[0] + agb2[0];
    const float gate = 1.0f / (1.0f + expf(-zz));
    const float aw = 1.0f / (1.0f + expf(-alpha[0]));
    const size_t r = (size_t)b * N + j;
    wsScale[r] = gate * aw + (1.0f - aw);
    wsRowBase[r] = (unsigned long long)base;
  }
}

// =====================================================================
// Kernel 4: QKV GEMM via fp32 WMMA (V_WMMA_F32_16X16X4_F32).
// Block = 128 threads (4 waves). Block tile 64(M) x 64(N); each wave owns
// 32x32 -> 2 A-frags x 2 B-frags -> 4 WMMA per K-step (8 FLOP/LDS-byte).
// Double-buffered LDS; staging via GLOBAL_LOAD_ASYNC_TO_LDS_B128 +
// s_wait_asynccnt (ASYNCcnt pipeline), sync fallback otherwise.
// Gate scale folded into epilogue: (s .* A) @ W == s .* (A @ W).
// =====================================================================
__global__ __launch_bounds__(128) void k_gemm_qkv(
    const float* __restrict__ tokens,
    const float* __restrict__ wq, const float* __restrict__ wk,
    const float* __restrict__ wvw,
    const int* __restrict__ wsN,
    const float* __restrict__ wsScale,
    const unsigned long long* __restrict__ wsRowBase,
    float* __restrict__ dout)
{
  const int N = wsN[0];
  const size_t Mrows = (size_t)B_DIM * N;
  const size_t Mtile0 = (size_t)blockIdx.y * 64;
  if (Mtile0 >= Mrows) return;        // block-uniform: EXEC stays all-ones
  const int Ntile0 = blockIdx.x * 64;
  const int z = blockIdx.z;
  const float* W = (z == 0) ? wq : (z == 1) ? wk : wvw;

  // A: 64 rows x 32 K, stride 36 -> b64 frag reads 8B-aligned, conflict-free
  // B: 32 K x 64 N, stride 72 -> lane-half windows hit disjoint banks
  __shared__ float As[2][64][36];
  __shared__ float Bs[2][32][72];

  const int tid = threadIdx.x;
  const int lane = tid & 31;
  const int wid = tid >> 5;
  const int ln = lane & 15;
  const int h = lane >> 4;
  const int mw = (wid & 1) * 32;      // wave M origin within block tile
  const int nw = (wid >> 1) * 32;     // wave N origin within block tile

  // staging assignments: A -> (row, 16-col half), B -> (k-row, 16-col quarter)
  const int alr = tid >> 1;           // 0..63
  const int alc = (tid & 1) * 16;
  const int bkr = tid >> 2;           // 0..31
  const int bnc = (tid & 3) * 16;

  const size_t ar = Mtile0 + alr;
  // rows past Mrows: clamp to a valid row; results land in M rows we never
  // store, so garbage there is harmless (no zero-fill needed).
  const size_t arc = (ar < Mrows) ? ar : (Mrows - 1);
  const unsigned long long abase = wsRowBase[arc];

  v8f acc00 = {}, acc01 = {}, acc10 = {}, acc11 = {};

  auto stage = [&](int k0, int bf) {
    const float* gA = tokens + (size_t)abase + (size_t)(k0 + alc);
    float* lA = &As[bf][alr][alc];
    const float* gB = W + (size_t)(k0 + bkr) * EXPERT_DIM + Ntile0 + bnc;
    float* lB = &Bs[bf][bkr][bnc];
#if HAVE_ASYNC_LDS
#pragma unroll
    for (int u = 0; u < 4; ++u) GLDS_B128(gA + u * 4, lA + u * 4);
#pragma unroll
    for (int u = 0; u < 4; ++u) GLDS_B128(gB + u * 4, lB + u * 4);
#else
#pragma unroll
    for (int u = 0; u < 16; ++u) lA[u] = gA[u];
#pragma unroll
    for (int u = 0; u < 16; ++u) lB[u] = gB[u];
#endif
  };
  auto wait_stage = [&]() {
#if HAVE_ASYNC_LDS
    __builtin_amdgcn_s_wait_asynccnt(0);
#endif
    __syncthreads();
  };

  int buf = 0;
  stage(0, buf);
  wait_stage();

  for (int k0 = 0; k0 < SLICE; k0 += 32) {
    if (k0 + 32 < SLICE) stage(k0 + 32, buf ^ 1);   // prefetch next chunk

#pragma unroll
    for (int ks = 0; ks < 32; ks += 4) {
      // A 16x4 frag (ISA 7.12.2): lane m = ln; vgpr0/1 hold K = 2h, 2h+1
      v2f a0, a1, b0, b1;
      a0.x = As[buf][mw + ln][ks + 2 * h];
      a0.y = As[buf][mw + ln][ks + 2 * h + 1];
      a1.x = As[buf][mw + 16 + ln][ks + 2 * h];
      a1.y = As[buf][mw + 16 + ln][ks + 2 * h + 1];
      // B 4x16 frag: N striped across lanes; symmetric K = 2h + vgpr
      b0.x = Bs[buf][ks + 2 * h][nw + ln];
      b0.y = Bs[buf][ks + 2 * h + 1][nw + ln];
      b1.x = Bs[buf][ks + 2 * h][nw + 16 + ln];
      b1.y = Bs[buf][ks + 2 * h + 1][nw + 16 + ln];
      acc00 = __builtin_amdgcn_wmma_f32_16x16x4_f32(
          false, a0, false, b0, (short)0, acc00, false, false);
      acc01 = __builtin_amdgcn_wmma_f32_16x16x4_f32(
          false, a0, false, b1, (short)0, acc01, false, false);
      acc10 = __builtin_amdgcn_wmma_f32_16x16x4_f32(
          false, a1, false, b0, (short)0, acc10, false, false);
      acc11 = __builtin_amdgcn_wmma_f32_16x16x4_f32(
          false, a1, false, b1, (short)0, acc11, false, false);
    }
    wait_stage();   // next chunk landed; also orders re-use of old buffer
    buf ^= 1;
  }

  // C/D layout: vgpr i -> M = i + 8*h, N = ln. Apply gate scale here.
  float* outBase = dout + (size_t)z * Mrows * EXPERT_DIM;
#pragma unroll
  for (int i = 0; i < 8; ++i) {
    const size_t r0 = Mtile0 + mw + i + 8 * h;
    const size_t r1 = r0 + 16;
    if (r0 < Mrows) {
      const float s0 = wsScale[r0];
      const size_t rowoff = r0 * EXPERT_DIM + Ntile0;
      outBase[rowoff + nw + ln] = acc00[i] * s0;
      outBase[rowoff + nw + 16 + ln] = acc01[i] * s0;
    }
    if (r1 < Mrows) {
      const float s1 = wsScale[r1];
      const size_t rowoff = r1 * EXPERT_DIM + Ntile0;
      outBase[rowoff + nw + ln] = acc10[i] * s1;
      outBase[rowoff + nw + 16 + ln] = acc11[i] * s1;
    }
  }
}

// =====================================================================
// Kernel 5: proj[t][v][b][j] = <qkv[t,b,j,:], dirs[v,:]>, t in (K,Q,V)
// =====================================================================
__global__ __launch_bounds__(128) void k_proj(
    const int* __restrict__ wsN, const float* __restrict__ wsDirs,
    float* __restrict__ dout)
{
  const int N = wsN[0];
  const int j = blockIdx.x;
  if (j >= N) return;
  const int b = blockIdx.y;
  const int t = blockIdx.z;                        // 0=K, 1=Q, 2=V
  const size_t Mrows = (size_t)B_DIM * N;
  const size_t r = (size_t)b * N + j;
  const int src = (t == 0) ? 1 : (t == 1) ? 0 : 2;
  const float* row = dout + (size_t)src * Mrows * EXPERT_DIM + r * EXPERT_DIM;

  const int tid = threadIdx.x;
  float acc[5] = {0, 0, 0, 0, 0};
  for (int k = tid; k < EXPERT_DIM; k += 128) {
    const float x = row[k];
#pragma unroll
    for (int v = 0; v < 5; ++v) acc[v] += x * wsDirs[v * EXPERT_DIM + k];
  }
  __shared__ float red[5][128];
#pragma unroll
  for (int v = 0; v < 5; ++v) red[v][tid] = acc[v];
  __syncthreads();
  for (int s = 64; s > 0; s >>= 1) {
    if (tid < s) {
#pragma unroll
      for (int v = 0; v < 5; ++v) red[v][tid] += red[v][tid + s];
    }
    __syncthreads();
  }
  if (tid < 5) {
    const size_t projBase = Mrows * (size_t)(3 * EXPERT_DIM);
    dout[projBase + (size_t)(t * 5 + tid) * Mrows + r] = red[tid][0];
  }
}

// =====================================================================
// Host launch
// =====================================================================
extern "C" void kernel_launch(void* const* d_in, const int* in_sizes, int n_in,
                              void* d_out, int out_size, void* d_ws,
                              size_t ws_size, hipStream_t stream)
{
  (void)in_sizes; (void)n_in; (void)out_size; (void)ws_size;
  const float* tokens = (const float*)d_in[0];
  const float* fp     = (const float*)d_in[1];
  const float* alpha  = (const float*)d_in[2];
  const float* agw1   = (const float*)d_in[3];
  const float* agb1   = (const float*)d_in[4];
  const float* agw2   = (const float*)d_in[5];
  const float* agb2   = (const float*)d_in[6];
  const float* wq     = (const float*)d_in[7];
  const float* wk     = (const float*)d_in[8];
  const float* wvw    = (const float*)d_in[9];
  const float* pent   = (const float*)d_in[10];
  float* dout = (float*)d_out;

  int* wsN   = (int*)d_ws;                                  // [0..15] N + pad
  int* wsIdx = wsN + 16;                                    // 4096 ints
  float* wsScale = (float*)(wsIdx + P_DIM);                 // 32768 f32
  unsigned long long* wsRowBase =
      (unsigned long long*)(wsScale + B_DIM * P_DIM);       // 32768 u64
  float* wsDirs = (float*)(wsRowBase + B_DIM * P_DIM);      // 5*2048 f32

  k_mask_compact<<<1, 1024, 0, stream>>>(fp, wsN, wsIdx, dout);
  k_dirs<<<5, 256, 0, stream>>>(pent, wsDirs);
  k_gate<<<dim3(P_DIM, B_DIM), 256, 0, stream>>>(
      tokens, agw1, agb1, agw2, agb2, alpha, wsN, wsIdx, wsScale, wsRowBase);
  k_gemm_qkv<<<dim3(EXPERT_DIM / 64, (B_DIM * P_DIM) / 64, 3), 128, 0, stream>>>(
      tokens, wq, wk, wvw, wsN, wsScale, wsRowBase, dout);
  k_proj<<<dim3(P_DIM, B_DIM, 3), 128, 0, stream>>>(wsN, wsDirs, dout);
}